// TwoLevelUNetWarp_64785286693673
// MI455X (gfx1250) — compile-verified
//
#include <hip/hip_runtime.h>

typedef __attribute__((ext_vector_type(2))) float v2f;
typedef __attribute__((ext_vector_type(8))) float v8f;

#define EPSV 1e-5f

// ---------------------------------------------------------------------------
// Repack weights W[oct][Cin][Cout] into per-lane WMMA B-fragment order:
//   R[(((wt*8+oct)*(Cin/4)+ks)*32+lane)*2 + j] = W[oct][ks*4 + 2*(lane>>4) + j][wt*16 + (lane&15)]
// so each wave's B operand per wmma step is one contiguous 256B chunk.
// ---------------------------------------------------------------------------
__global__ void repack_w(const float* __restrict__ W, float* __restrict__ R,
                         int Cin, int Cout)
{
    int total = (Cout / 16) * 8 * (Cin / 4) * 64;   // *32 lanes *2 elems
    for (int o = blockIdx.x * blockDim.x + threadIdx.x; o < total;
         o += gridDim.x * blockDim.x) {
        int j    = o & 1;
        int lane = (o >> 1) & 31;
        int rest = o >> 6;
        int ks   = rest % (Cin / 4);
        rest    /= (Cin / 4);
        int oct  = rest & 7;
        int wt   = rest >> 3;
        int c = ks * 4 + ((lane >> 4) << 1) + j;
        int n = wt * 16 + (lane & 15);
        R[o] = W[(oct * Cin + c) * Cout + n];
    }
}

// ---------------------------------------------------------------------------
// Sparse stride-2 conv (gather form): Y[m,:] = sum_k mask[m,k] * X[idx[m,k],:] @ W[k]
// Block: 16 output rows, (COUT/16) waves; wave w computes columns [16w,16w+16).
// ---------------------------------------------------------------------------
template <int CIN, int COUT>
__global__ __launch_bounds__(32 * (COUT / 16))
void conv_gather(const float* __restrict__ X, const int* __restrict__ idx,
                 const unsigned char* __restrict__ mask,
                 const float* __restrict__ Wre, float* __restrict__ Y, int M)
{
    constexpr int NT  = COUT / 16;       // waves per block
    constexpr int ROW = 8 * CIN + 4;     // LDS row stride (floats), bank-conflict pad
    __shared__ float ldsA[16 * ROW];

    const int t    = threadIdx.x;
    const int lane = t & 31;
    const int wave = t >> 5;
    const int m0   = blockIdx.x * 16;

    // cooperative masked gather: 16 rows x 8 octants x CIN floats (float4 chunks)
    constexpr int NV4 = 16 * 8 * CIN / 4;
    for (int e = t; e < NV4; e += 32 * NT) {
        int c4   = e % (CIN / 4);
        int rest = e / (CIN / 4);
        int oct  = rest & 7;
        int m    = rest >> 3;
        int gi   = idx[(m0 + m) * 8 + oct];
        float4 v = {0.f, 0.f, 0.f, 0.f};
        if (mask[(m0 + m) * 8 + oct])
            v = *(const float4*)(X + (long)gi * CIN + c4 * 4);
        *(float4*)&ldsA[m * ROW + oct * CIN + c4 * 4] = v;
    }
    __syncthreads();

    const int mr = lane & 15;
    const int kh = (lane >> 4) << 1;     // 0 or 2: K-half per WMMA f32 layout
    v8f acc = {};

    for (int oct = 0; oct < 8; ++oct) {
        const float* arow = &ldsA[mr * ROW + oct * CIN + kh];
        const float* brow = Wre + ((wave * 8 + oct) * (CIN / 4) * 32 + lane) * 2;
#pragma unroll
        for (int ks = 0; ks < CIN / 4; ++ks) {
            v2f a = *(const v2f*)(arow + ks * 4);
            v2f b = *(const v2f*)(brow + ks * 64);
            acc = __builtin_amdgcn_wmma_f32_16x16x4_f32(
                false, a, false, b, (short)0, acc, false, false);
        }
    }

    const int col = wave * 16 + mr;
#pragma unroll
    for (int r = 0; r < 8; ++r) {
        int row = m0 + r + ((lane >> 4) << 3);
        Y[(long)row * COUT + col] = acc[r];
    }
}

// ---------------------------------------------------------------------------
// Transposed stride-2 conv (scatter form): for each parent p and octant o,
//   if mask[p,o]: Y[idx[p,o],:] = F[p,:] @ W[o]
// Block: 16 parent rows, (COUT/16) waves.
// ---------------------------------------------------------------------------
template <int CIN, int COUT>
__global__ __launch_bounds__(32 * (COUT / 16))
void conv_scatter(const float* __restrict__ F, const int* __restrict__ idx,
                  const unsigned char* __restrict__ mask,
                  const float* __restrict__ Wre, float* __restrict__ Y, int P)
{
    constexpr int NT  = COUT / 16;
    constexpr int ROW = CIN + 4;
    __shared__ float ldsF[16 * ROW];

    const int t    = threadIdx.x;
    const int lane = t & 31;
    const int wave = t >> 5;
    const int p0   = blockIdx.x * 16;

    for (int e = t; e < 16 * CIN / 4; e += 32 * NT) {
        int c4 = e % (CIN / 4);
        int m  = e / (CIN / 4);
        *(float4*)&ldsF[m * ROW + c4 * 4] =
            *(const float4*)(F + (long)(p0 + m) * CIN + c4 * 4);
    }
    __syncthreads();

    const int mr = lane & 15;
    const int kh = (lane >> 4) << 1;

    for (int oct = 0; oct < 8; ++oct) {
        v8f acc = {};
        const float* arow = &ldsF[mr * ROW + kh];
        const float* brow = Wre + ((wave * 8 + oct) * (CIN / 4) * 32 + lane) * 2;
#pragma unroll
        for (int ks = 0; ks < CIN / 4; ++ks) {
            v2f a = *(const v2f*)(arow + ks * 4);
            v2f b = *(const v2f*)(brow + ks * 64);
            acc = __builtin_amdgcn_wmma_f32_16x16x4_f32(
                false, a, false, b, (short)0, acc, false, false);
        }
        const int col = wave * 16 + mr;
#pragma unroll
        for (int r = 0; r < 8; ++r) {
            int prow = p0 + r + ((lane >> 4) << 3);
            int child = idx[prow * 8 + oct];
            if (mask[prow * 8 + oct])
                Y[(long)child * COUT + col] = acc[r];
        }
    }
}

// ---------------------------------------------------------------------------
// Instance-norm statistics: deterministic two-pass reduction.
// ---------------------------------------------------------------------------
template <int C>
__global__ void rstats_partial(const float* __restrict__ Y, float* __restrict__ part,
                               int M)
{
    __shared__ float ls[256], ls2[256];
    const int t = threadIdx.x;
    const int c = t & (C - 1);
    const int g = t / C;
    constexpr int G = 256 / C;
    float s = 0.f, s2 = 0.f;
    for (long r = (long)blockIdx.x * G + g; r < M; r += (long)gridDim.x * G) {
        float v = Y[r * C + c];
        s += v; s2 += v * v;
    }
    ls[t] = s; ls2[t] = s2;
    __syncthreads();
    if (t < C) {
        float S = 0.f, S2 = 0.f;
#pragma unroll
        for (int i = 0; i < G; ++i) { S += ls[i * C + t]; S2 += ls2[i * C + t]; }
        part[blockIdx.x * 2 * C + t]     = S;
        part[blockIdx.x * 2 * C + C + t] = S2;
    }
}

template <int C>
__global__ void rstats_final(const float* __restrict__ part,
                             const float* __restrict__ g, const float* __restrict__ b,
                             float* __restrict__ ab, int NB, int M)
{
    int c = threadIdx.x;
    if (c < C) {
        float S = 0.f, S2 = 0.f;
        for (int i = 0; i < NB; ++i) {
            S  += part[i * 2 * C + c];
            S2 += part[i * 2 * C + C + c];
        }
        float mean = S / (float)M;
        float var  = S2 / (float)M - mean * mean;
        float a    = g[c] * rsqrtf(var + EPSV);
        ab[c]      = a;
        ab[C + c]  = b[c] - mean * a;
    }
}

template <int C>
__global__ void normalize_ip(float* __restrict__ Y, const float* __restrict__ ab,
                             long M)
{
    long n = M * C;
    for (long i = (long)blockIdx.x * blockDim.x + threadIdx.x; i < n;
         i += (long)gridDim.x * blockDim.x) {
        int c = (int)(i & (C - 1));
        Y[i] = fmaf(ab[c], Y[i], ab[C + c]);
    }
}

// ---------------------------------------------------------------------------
extern "C" void kernel_launch(void* const* d_in, const int* in_sizes, int n_in,
                              void* d_out, int out_size, void* d_ws, size_t ws_size,
                              hipStream_t stream)
{
    const float* x     = (const float*)d_in[0];
    const float* W_up1 = (const float*)d_in[1];
    const float* W_up2 = (const float*)d_in[2];
    const float* W_d2  = (const float*)d_in[3];
    const float* W_d1  = (const float*)d_in[4];
    const float* g1 = (const float*)d_in[5];  const float* b1 = (const float*)d_in[6];
    const float* g2 = (const float*)d_in[7];  const float* b2 = (const float*)d_in[8];
    const float* g3 = (const float*)d_in[9];  const float* b3 = (const float*)d_in[10];
    const float* g4 = (const float*)d_in[11]; const float* b4 = (const float*)d_in[12];
    const int* idx_up1 = (const int*)d_in[13];
    const unsigned char* mask_up1 = (const unsigned char*)d_in[14];
    const int* idx_up2 = (const int*)d_in[15];
    const unsigned char* mask_up2 = (const unsigned char*)d_in[16];
    // d_in[17..20] (par/off) are unneeded: idx_up*/mask_up* are their inverse maps.

    const int N0 = 800000, N1 = 200000, N2 = 32768;
    const int NB = 240;

    float* out    = (float*)d_out;
    float* xdown2 = out;                        // [N0,32]
    float* xdown1 = out + (long)N0 * 32;        // [N1,64]
    float* xup2   = xdown1 + (long)N1 * 64;     // [N2,128]
    float* xup1   = xup2 + (long)N2 * 128;      // [N1,64]

    float* w        = (float*)d_ws;
    float* Wre_up1  = w;                    // 4*8*8*64   = 16384
    float* Wre_up2  = Wre_up1 + 16384;      // 8*8*16*64  = 65536
    float* Wre_d2   = Wre_up2 + 65536;      // 4*8*32*64  = 65536
    float* Wre_d1   = Wre_d2  + 65536;      // 2*8*16*64  = 16384
    float* part     = Wre_d1  + 16384;      // NB*2*128   = 61440
    float* ab       = part + NB * 256;      // 256

    repack_w<<<64,  256, 0, stream>>>(W_up1, Wre_up1, 32, 64);
    repack_w<<<256, 256, 0, stream>>>(W_up2, Wre_up2, 64, 128);
    repack_w<<<256, 256, 0, stream>>>(W_d2,  Wre_d2,  128, 64);
    repack_w<<<64,  256, 0, stream>>>(W_d1,  Wre_d1,  64, 32);

    // Stage 1: x_up1 = inorm(sconv(x, idx_up1, mask_up1, W_up1))
    conv_gather<32, 64><<<N1 / 16, 128, 0, stream>>>(x, idx_up1, mask_up1, Wre_up1, xup1, N1);
    rstats_partial<64><<<NB, 256, 0, stream>>>(xup1, part, N1);
    rstats_final<64><<<1, 128, 0, stream>>>(part, g1, b1, ab, NB, N1);
    normalize_ip<64><<<2048, 256, 0, stream>>>(xup1, ab, N1);

    // Stage 2: x_up2 = inorm(sconv(x_up1, idx_up2, mask_up2, W_up2))
    conv_gather<64, 128><<<N2 / 16, 256, 0, stream>>>(xup1, idx_up2, mask_up2, Wre_up2, xup2, N2);
    rstats_partial<128><<<NB, 256, 0, stream>>>(xup2, part, N2);
    rstats_final<128><<<1, 128, 0, stream>>>(part, g2, b2, ab, NB, N2);
    normalize_ip<128><<<2048, 256, 0, stream>>>(xup2, ab, N2);

    // Stage 3: x_down1 = inorm(tconv(x_up2, par_d2, off_d2, W_d2))  — scatter form
    conv_scatter<128, 64><<<N2 / 16, 128, 0, stream>>>(xup2, idx_up2, mask_up2, Wre_d2, xdown1, N2);
    rstats_partial<64><<<NB, 256, 0, stream>>>(xdown1, part, N1);
    rstats_final<64><<<1, 128, 0, stream>>>(part, g3, b3, ab, NB, N1);
    normalize_ip<64><<<2048, 256, 0, stream>>>(xdown1, ab, N1);

    // Stage 4: x_down2 = inorm(tconv(x_down1, par_d1, off_d1, W_d1)) — scatter form
    conv_scatter<64, 32><<<N1 / 16, 64, 0, stream>>>(xdown1, idx_up1, mask_up1, Wre_d1, xdown2, N1);
    rstats_partial<32><<<NB, 256, 0, stream>>>(xdown2, part, N0);
    rstats_final<32><<<1, 128, 0, stream>>>(part, g4, b4, ab, NB, N0);
    normalize_ip<32><<<2048, 256, 0, stream>>>(xdown2, ab, N0);
}